// TDVP_PS_79577154060545
// MI455X (gfx1250) — compile-verified
//
#include <hip/hip_runtime.h>

#define BS_ 16384
#define N_  256

typedef float v2f __attribute__((ext_vector_type(2)));
typedef float v8f __attribute__((ext_vector_type(8)));

__device__ __forceinline__ float clipf(float v) {
  return fminf(1000.0f, fmaxf(-1000.0f, v));
}

// Deterministic block reduction (blockDim.x == 256).
__device__ __forceinline__ float block_reduce_256(float v, float* sm) {
  __syncthreads();
  const int t = threadIdx.x;
  sm[t] = v;
  __syncthreads();
#pragma unroll
  for (int s = 128; s > 0; s >>= 1) {
    if (t < s) sm[t] += sm[t + s];
    __syncthreads();
  }
  return sm[0];
}

// Level offsets (float4 units) for levels 0..7: level t holds (256>>t)*BS mats.
constexpr unsigned OFF4[8] = {0u,       4194304u, 6291456u, 7340032u,
                              7864320u, 8126464u, 8257536u, 8323072u};

// ---------------------------------------------------------------------------
// Stage 1: mpoc[row, lr] = sum_ij (x_i x_j) * mpo_m[lr, ij]   via WMMA f32 16x16x4
//   A (16x4) = mpo_m as (lr x ij), row lr==2 masked (triu mask), rows 4..15 zero
//   B (4x16) = XX^T, column N = wave row base+N (WMMA1) / base+16+N (WMMA2)
//   D row m = lr, column N = row  ->  lane N holds mpoc[row, 0..3] in c[0..3]
// ---------------------------------------------------------------------------
__global__ __launch_bounds__(256) void k_mpoc(const float* __restrict__ x,
                                              const float* __restrict__ mpo,
                                              float* __restrict__ lvl0) {
  const int tid      = blockIdx.x * 256 + threadIdx.x;  // flat row n*BS+b
  const int lane     = threadIdx.x & 31;
  const int waveBase = tid - lane;

  const float2 xv = ((const float2*)x)[tid];

  const int  mrow = lane & 15;
  const bool hiK  = lane >= 16;  // lanes 16-31 carry K=2,3 of A / B
  v2f A;
  {
    // Unconditional (index-clamped) loads -> no exec-mask churn before WMMA.
    const bool rowOk = (mrow < 4) && (mrow != 2);  // lr==2 masked by triu
    const int  kk    = hiK ? 2 : 0;
    const float v0 = mpo[(mrow & 3) * 4 + kk];
    const float v1 = mpo[(mrow & 3) * 4 + kk + 1];
    A[0] = rowOk ? v0 : 0.0f;
    A[1] = rowOk ? v1 : 0.0f;
  }

  v8f c1 = {}; v8f c2 = {};
  {
    const int   src = lane & 15;            // rows base+0..15
    const float sx0 = __shfl(xv.x, src, 32);
    const float sx1 = __shfl(xv.y, src, 32);
    v2f B;
    B[0] = hiK ? sx1 * sx0 : sx0 * sx0;     // K=2: x1x0 | K=0: x0x0
    B[1] = hiK ? sx1 * sx1 : sx0 * sx1;     // K=3: x1x1 | K=1: x0x1
    c1 = __builtin_amdgcn_wmma_f32_16x16x4_f32(false, A, false, B, (short)0, c1,
                                               false, false);
  }
  {
    const int   src = 16 + (lane & 15);     // rows base+16..31
    const float sx0 = __shfl(xv.x, src, 32);
    const float sx1 = __shfl(xv.y, src, 32);
    v2f B;
    B[0] = hiK ? sx1 * sx0 : sx0 * sx0;
    B[1] = hiK ? sx1 * sx1 : sx0 * sx1;
    c2 = __builtin_amdgcn_wmma_f32_16x16x4_f32(false, A, false, B, (short)0, c2,
                                               false, false);
  }
  if (lane < 16) {
    ((float4*)lvl0)[waveBase + lane]      = make_float4(c1[0], c1[1], c1[2], c1[3]);
    ((float4*)lvl0)[waveBase + 16 + lane] = make_float4(c2[0], c2[1], c2[2], c2[3]);
  }
}

// ---------------------------------------------------------------------------
// Stage 2 (fused): one tree level, grid-stride over all pairs with a fixed
// 512-block grid. Each block first redundantly recomputes the PREVIOUS level's
// global sum-of-squares from its 512 block-partials (deterministic, identical
// in all blocks), so no separate reduce kernel is needed between levels.
//   out[p] = (in[2p] x in[2p+1]) / ssq_prev ;  partials of new ssq emitted.
// ---------------------------------------------------------------------------
__global__ __launch_bounds__(256) void k_pair(const float* __restrict__ in,
                                              float* __restrict__ outl,
                                              const float* __restrict__ prevPart,
                                              float* __restrict__ partial,
                                              int total) {
  __shared__ float sm[256];

  float s = 1.0f;  // invg_prev^2
  if (prevPart != nullptr) {
    float acc = prevPart[threadIdx.x] + prevPart[threadIdx.x + 256];
    const float tot = block_reduce_256(acc, sm);
    s = 1.0f / tot;
  }

  const size_t stride = (size_t)gridDim.x * 256;
  float ssq = 0.0f;
  for (size_t tid = (size_t)blockIdx.x * 256 + threadIdx.x; tid < (size_t)total;
       tid += stride) {
    const int p = (int)(tid >> 14);
    const int b = (int)(tid & (BS_ - 1));

    // Prefetch next grid-stride iteration's inputs (global_prefetch_b8).
    const size_t nx = tid + stride;
    if (nx < (size_t)total) {
      const int pn = (int)(nx >> 14);
      const int bn = (int)(nx & (BS_ - 1));
      __builtin_prefetch(&((const float4*)in)[(size_t)(2 * pn) * BS_ + bn], 0, 0);
      __builtin_prefetch(&((const float4*)in)[(size_t)(2 * pn + 1) * BS_ + bn], 0, 0);
    }

    const float4 A4 = ((const float4*)in)[(size_t)(2 * p)     * BS_ + b];
    const float4 B4 = ((const float4*)in)[(size_t)(2 * p + 1) * BS_ + b];
    const float c0 = (A4.x * B4.x + A4.y * B4.z) * s;
    const float c1 = (A4.x * B4.y + A4.y * B4.w) * s;
    const float c2 = (A4.z * B4.x + A4.w * B4.z) * s;
    const float c3 = (A4.z * B4.y + A4.w * B4.w) * s;
    ((float4*)outl)[(size_t)p * BS_ + b] = make_float4(c0, c1, c2, c3);
    ssq += c0 * c0 + c1 * c1 + c2 * c2 + c3 * c3;
  }

  const float tot = block_reduce_256(ssq, sm);
  if (threadIdx.x == 0) partial[blockIdx.x] = tot;
}

// Finalize gbuf[t] = total ssq of level t (t = 1..7) for the Al/Ar walk.
__global__ __launch_bounds__(256) void k_reduce_all(const float* __restrict__ parts,
                                                    float* __restrict__ gbuf) {
  __shared__ float sm[256];
  const int t = blockIdx.x + 1;
  const float* p = parts + (size_t)t * 512;
  float acc = p[threadIdx.x] + p[threadIdx.x + 256];
  const float tot = block_reduce_256(acc, sm);
  if (threadIdx.x == 0) gbuf[t] = tot;
}

// ---------------------------------------------------------------------------
// Stage 3: Al/Ar tree walk. n = blockIdx.y is scalar per block. Explicit
// load phase (addresses depend only on (n,b)) issues all gathers back-to-back
// for MLP; the levels array (~128MB) is L2-resident (192MB L2).
// ---------------------------------------------------------------------------
__global__ __launch_bounds__(256) void k_alar(const float* __restrict__ lv,
                                              const float* __restrict__ gbuf,
                                              float* __restrict__ AlBuf,
                                              float* __restrict__ ArBuf,
                                              float* __restrict__ partialL,
                                              float* __restrict__ partialR) {
  __shared__ float sm[256];
  const int n = blockIdx.y;
  const int b = blockIdx.x * 256 + threadIdx.x;

  float ig[8];
  ig[0] = 1.0f;  // level 0 (mpoc) is not normalized
#pragma unroll
  for (int t = 1; t < 8; ++t) ig[t] = rsqrtf(gbuf[t]);

  // ---- load phase: gather every needed matrix first ----
  float4 ml[8], mr[8];
#pragma unroll
  for (int j = 0; j < 8; ++j) {
    const int lev = 7 - j;
    if (((n >> lev) & 1) == 0) continue;
    const int half = n >> (lev + 1);
    const int idxl = 2 * half;
    const int sz   = 1 << (8 - lev);
    const int idxr = sz - 1 - 2 * half;
    ml[lev] = ((const float4*)lv)[(size_t)OFF4[lev] + (size_t)idxl * BS_ + b];
    mr[lev] = ((const float4*)lv)[(size_t)OFF4[lev] + (size_t)idxr * BS_ + b];
  }

  // ---- combine phase ----
  float al0 = 0.f, al1 = 0.f, ar0 = 0.f, ar1 = 0.f;
  bool started = false;
#pragma unroll
  for (int j = 0; j < 8; ++j) {
    const int lev = 7 - j;
    if (((n >> lev) & 1) == 0) continue;
    const float g = ig[lev];
    {  // left environment: start = row 0, cont = rowvec * M, clipped
      const float m0 = ml[lev].x * g, m1 = ml[lev].y * g;
      const float m2 = ml[lev].z * g, m3 = ml[lev].w * g;
      if (!started) { al0 = m0; al1 = m1; }
      else {
        const float t0 = al0 * m0 + al1 * m2;
        const float t1 = al0 * m1 + al1 * m3;
        al0 = clipf(t0); al1 = clipf(t1);
      }
    }
    {  // right environment: start = last column, cont = M * colvec, clipped
      const float m0 = mr[lev].x * g, m1 = mr[lev].y * g;
      const float m2 = mr[lev].z * g, m3 = mr[lev].w * g;
      if (!started) { ar0 = m1; ar1 = m3; }
      else {
        const float t0 = m0 * ar0 + m1 * ar1;
        const float t1 = m2 * ar0 + m3 * ar1;
        ar0 = clipf(t0); ar1 = clipf(t1);
      }
    }
    started = true;
  }

  const size_t row = (size_t)n * BS_ + b;
  ((float2*)AlBuf)[row] = make_float2(al0, al1);
  ((float2*)ArBuf)[row] = make_float2(ar0, ar1);

  const float tl = block_reduce_256(al0 * al0 + al1 * al1, sm);
  if (threadIdx.x == 0) partialL[n * 64 + blockIdx.x] = tl;
  const float tr = block_reduce_256(ar0 * ar0 + ar1 * ar1, sm);
  if (threadIdx.x == 0) partialR[n * 64 + blockIdx.x] = tr;
}

__global__ __launch_bounds__(64) void k_norm(const float* __restrict__ pL,
                                             const float* __restrict__ pR,
                                             float* __restrict__ invnl,
                                             float* __restrict__ invnr) {
  __shared__ float sm[64];
  const int n = blockIdx.x;
  const int t = threadIdx.x;
  sm[t] = pL[n * 64 + t];
  __syncthreads();
#pragma unroll
  for (int s = 32; s > 0; s >>= 1) { if (t < s) sm[t] += sm[t + s]; __syncthreads(); }
  if (t == 0) invnl[n] = rsqrtf(sm[0]);
  __syncthreads();
  sm[t] = pR[n * 64 + t];
  __syncthreads();
#pragma unroll
  for (int s = 32; s > 0; s >>= 1) { if (t < s) sm[t] += sm[t + s]; __syncthreads(); }
  if (t == 0) invnr[n] = rsqrtf(sm[0]);
}

// ---------------------------------------------------------------------------
// Stage 4: H[row, ij] = sum_lr (Al_l Ar_r)[row] * scale*mpo_m[lr, ij]  via WMMA,
// then out = leaky(H x) / rowsum. Same A/B orientation as k_mpoc.
// ---------------------------------------------------------------------------
__global__ __launch_bounds__(256) void k_out(const float* __restrict__ x,
                                             const float* __restrict__ mpo,
                                             const float* __restrict__ Bl,
                                             const float* __restrict__ Br,
                                             const float* __restrict__ scalep,
                                             const float* __restrict__ AlBuf,
                                             const float* __restrict__ ArBuf,
                                             const float* __restrict__ invnl,
                                             const float* __restrict__ invnr,
                                             float* __restrict__ outp) {
  const int    n    = blockIdx.y;
  const int    b    = blockIdx.x * 256 + threadIdx.x;
  const int    lane = threadIdx.x & 31;
  const size_t row  = (size_t)n * BS_ + b;
  const float scale = scalep[0];

  float al0, al1, ar0, ar1;
  if (n == 0) { al0 = Bl[0]; al1 = Bl[1]; }
  else {
    const float il = invnl[n];
    al0 = AlBuf[row * 2 + 0] * il;
    al1 = AlBuf[row * 2 + 1] * il;
  }
  if (n == N_ - 1) { ar0 = Br[0]; ar1 = Br[1]; }
  else {
    const int   m  = N_ - 1 - n;
    const float ir = invnr[m];
    ar0 = ArBuf[((size_t)m * BS_ + b) * 2 + 0] * ir;
    ar1 = ArBuf[((size_t)m * BS_ + b) * 2 + 1] * ir;
  }
  const float k0 = al0 * ar0, k1 = al0 * ar1, k2 = al1 * ar0, k3 = al1 * ar1;

  // A = scale * mpo_m as (ij x lr); K column lr==2 masked by triu.
  const int  mrow = lane & 15;
  const bool hiK  = lane >= 16;
  v2f A;
  {
    const bool rowOk = mrow < 4;
    const int  ij    = mrow & 3;  // clamped index: unconditional loads
    const float vlo0 = mpo[0 + ij] * scale;    // lr = 0
    const float vlo1 = mpo[4 + ij] * scale;    // lr = 1
    const float vhi1 = mpo[12 + ij] * scale;   // lr = 3
    if (!hiK) {
      A[0] = rowOk ? vlo0 : 0.0f;
      A[1] = rowOk ? vlo1 : 0.0f;
    } else {
      A[0] = 0.0f;                             // lr = 2 (masked)
      A[1] = rowOk ? vhi1 : 0.0f;
    }
  }

  v8f c1 = {}; v8f c2 = {};
  {
    const int   src = lane & 15;
    const float t0 = __shfl(k0, src, 32);
    const float t1 = __shfl(k1, src, 32);
    const float t2 = __shfl(k2, src, 32);
    const float t3 = __shfl(k3, src, 32);
    v2f B; B[0] = hiK ? t2 : t0; B[1] = hiK ? t3 : t1;
    c1 = __builtin_amdgcn_wmma_f32_16x16x4_f32(false, A, false, B, (short)0, c1,
                                               false, false);
  }
  {
    const int   src = 16 + (lane & 15);
    const float t0 = __shfl(k0, src, 32);
    const float t1 = __shfl(k1, src, 32);
    const float t2 = __shfl(k2, src, 32);
    const float t3 = __shfl(k3, src, 32);
    v2f B; B[0] = hiK ? t2 : t0; B[1] = hiK ? t3 : t1;
    c2 = __builtin_amdgcn_wmma_f32_16x16x4_f32(false, A, false, B, (short)0, c2,
                                               false, false);
  }

  const float2 xv  = ((const float2*)x)[row];
  const float  xb0 = __shfl(xv.x, 16 + (lane & 15), 32);
  const float  xb1 = __shfl(xv.y, 16 + (lane & 15), 32);

  if (lane < 16) {
    const size_t rbase = row - lane;  // wave's first row
    float o0 = c1[0] * xv.x + c1[1] * xv.y;
    float o1 = c1[2] * xv.x + c1[3] * xv.y;
    o0 = o0 > 0.0f ? o0 : (-0.125f) * o0;
    o1 = o1 > 0.0f ? o1 : (-0.125f) * o1;
    const float s1 = o0 + o1;
    ((float2*)outp)[rbase + lane] = make_float2(o0 / s1, o1 / s1);

    float p0 = c2[0] * xb0 + c2[1] * xb1;
    float p1 = c2[2] * xb0 + c2[3] * xb1;
    p0 = p0 > 0.0f ? p0 : (-0.125f) * p0;
    p1 = p1 > 0.0f ? p1 : (-0.125f) * p1;
    const float s2 = p0 + p1;
    ((float2*)outp)[rbase + 16 + lane] = make_float2(p0 / s2, p1 / s2);
  }
}

// ---------------------------------------------------------------------------
extern "C" void kernel_launch(void* const* d_in, const int* in_sizes, int n_in,
                              void* d_out, int out_size, void* d_ws, size_t ws_size,
                              hipStream_t stream) {
  const float* x     = (const float*)d_in[0];
  const float* mpo   = (const float*)d_in[1];
  const float* Bl    = (const float*)d_in[2];
  const float* Br    = (const float*)d_in[3];
  const float* scale = (const float*)d_in[4];
  float* out = (float*)d_out;
  float* ws  = (float*)d_ws;

  // Workspace layout (floats). Levels 0..7 (level 8 of the reference is never
  // read by the Al/Ar walk, so it is skipped entirely).
  float* levels   = ws;                   // 33,423,360
  float* AlBuf    = levels + 33423360;    //  8,388,608
  float* ArBuf    = AlBuf + 8388608;      //  8,388,608
  float* parts    = ArBuf + 8388608;      //  8 * 512  (per-level block partials)
  float* gbuf     = parts + 8 * 512;      //  8        (ssq per level)
  float* partialL = gbuf + 8;             //  16,384
  float* partialR = partialL + 16384;     //  16,384
  float* invnl    = partialR + 16384;     //  256
  float* invnr    = invnl + 256;          //  256

  static const size_t loff[8] = {0,        16777216, 25165824, 29360128,
                                 31457280, 32505856, 33030144, 33292288};

  k_mpoc<<<16384, 256, 0, stream>>>(x, mpo, levels);

  for (int t = 1; t < 8; ++t) {
    const int total = (256 >> t) * BS_;
    k_pair<<<512, 256, 0, stream>>>(levels + loff[t - 1], levels + loff[t],
                                    (t == 1) ? (const float*)nullptr
                                             : (const float*)(parts + (size_t)(t - 1) * 512),
                                    parts + (size_t)t * 512, total);
  }
  k_reduce_all<<<7, 256, 0, stream>>>(parts, gbuf);

  k_alar<<<dim3(BS_ / 256, N_), 256, 0, stream>>>(levels, gbuf, AlBuf, ArBuf,
                                                  partialL, partialR);
  k_norm<<<N_, 64, 0, stream>>>(partialL, partialR, invnl, invnr);
  k_out<<<dim3(BS_ / 256, N_), 256, 0, stream>>>(x, mpo, Bl, Br, scale, AlBuf,
                                                 ArBuf, invnl, invnr, out);

  (void)in_sizes; (void)n_in; (void)out_size; (void)ws_size;
}